// RGCN_50405736186438
// MI455X (gfx1250) — compile-verified
//
#include <hip/hip_runtime.h>

typedef float v2f __attribute__((ext_vector_type(2)));
typedef float v8f __attribute__((ext_vector_type(8)));

#define N_NODES 40000
#define E_EDGES 640000
#define R_REL   8
#define H_DIM   128
#define C_DIM   64

// ---------------------------------------------------------------------------
// 1) zero workspace accumulators
// ---------------------------------------------------------------------------
__global__ void zero_kernel(float* __restrict__ p, int n) {
    int i = blockIdx.x * blockDim.x + threadIdx.x;
    if (i < n) p[i] = 0.0f;
}

// ---------------------------------------------------------------------------
// 2) per-(dst, rel) edge counts (for mean)
// ---------------------------------------------------------------------------
__global__ void count_kernel(const int* __restrict__ dst,
                             const int* __restrict__ et,
                             float* __restrict__ cnt) {
    int e = blockIdx.x * blockDim.x + threadIdx.x;
    if (e < E_EDGES) atomicAdd(&cnt[dst[e] * R_REL + et[e]], 1.0f);
}

// ---------------------------------------------------------------------------
// 3) conv1 message scatter: h[dst,:] += W1[rel, src, :] / cnt[dst,rel]
//    one block per edge, 128 threads = one feature each (coalesced 512B row)
// ---------------------------------------------------------------------------
__global__ __launch_bounds__(H_DIM) void scatter1_kernel(
        const int* __restrict__ src, const int* __restrict__ dst,
        const int* __restrict__ et,  const float* __restrict__ W1,
        const float* __restrict__ cnt, float* __restrict__ h) {
    int e = blockIdx.x;
    int f = threadIdx.x;
    int s = src[e], d = dst[e], r = et[e];
    float inv = 1.0f / cnt[d * R_REL + r];        // cnt >= 1 for this edge's segment
    float v = W1[((size_t)r * N_NODES + s) * H_DIM + f] * inv;
    atomicAdd(&h[(size_t)d * H_DIM + f], v);
}

// ---------------------------------------------------------------------------
// 4) h = relu(h + root1 + b1)
// ---------------------------------------------------------------------------
__global__ void relu_add_kernel(float* __restrict__ h,
                                const float* __restrict__ root1,
                                const float* __restrict__ b1) {
    int i = blockIdx.x * blockDim.x + threadIdx.x;
    if (i < N_NODES * H_DIM) {
        float v = h[i] + root1[i] + b1[i & (H_DIM - 1)];
        h[i] = v > 0.0f ? v : 0.0f;
    }
}

// ---------------------------------------------------------------------------
// 5) WMMA f32 GEMM:  z<8 : h_all[z] = h @ W2[z]   ([N,H]x[H,C])
//                    z==8: hroot    = h @ root2
//    One wave32 computes one 16x16 output tile; K-loop over H in steps of 4
//    using V_WMMA_F32_16X16X4_F32 with accumulator chaining.
//    grid = (625, 4, 9), block = 128 (4 waves -> 4 consecutive m-tiles);
//    2500 m-tiles = 625*4 exactly, so no divergence: EXEC stays all-ones.
// ---------------------------------------------------------------------------
__global__ __launch_bounds__(128) void wmma_gemm_kernel(
        const float* __restrict__ h,      // [N, H]
        const float* __restrict__ W2,     // [R, H, C]
        const float* __restrict__ root2,  // [H, C]
        float* __restrict__ h_all,        // [R, N, C]
        float* __restrict__ hroot) {      // [N, C]
    const int lane  = threadIdx.x & 31;
    const int wave  = threadIdx.x >> 5;
    const int mtile = blockIdx.x * 4 + wave;   // 0..2499
    const int ntile = blockIdx.y;              // 0..3
    const int z     = blockIdx.z;              // 0..8

    const float* __restrict__ B = (z < R_REL) ? (W2 + (size_t)z * H_DIM * C_DIM) : root2;
    float* __restrict__       D = (z < R_REL) ? (h_all + (size_t)z * N_NODES * C_DIM) : hroot;

    const int half = lane >> 4;                // 0: K offsets {0,1}; 1: {2,3}
    const int l15  = lane & 15;
    const int arow = mtile * 16 + l15;         // A-matrix row (node), per ISA layout
    const int bcol = ntile * 16 + l15;         // B-matrix column

    const float* __restrict__ arowp = h + (size_t)arow * H_DIM + 2 * half;

    v8f acc = {};
    #pragma unroll 4
    for (int k = 0; k < H_DIM; k += 4) {
        v2f a, b;
        a.x = arowp[k];
        a.y = arowp[k + 1];
        int kb = k + 2 * half;
        b.x = B[(size_t)kb * C_DIM + bcol];
        b.y = B[(size_t)(kb + 1) * C_DIM + bcol];
        acc = __builtin_amdgcn_wmma_f32_16x16x4_f32(
            /*neg_a=*/false, a, /*neg_b=*/false, b,
            /*c_mod=*/(short)0, acc, /*reuse_a=*/false, /*reuse_b=*/false);
    }

    // D layout: VGPR v holds row (v + 8*half), column l15
    #pragma unroll
    for (int v = 0; v < 8; ++v) {
        int m = mtile * 16 + v + 8 * half;
        D[(size_t)m * C_DIM + bcol] = acc[v];
    }
}

// ---------------------------------------------------------------------------
// 6) out = hroot + b2  (initializes d_out before the atomic scatter)
// ---------------------------------------------------------------------------
__global__ void init_out_kernel(const float* __restrict__ hroot,
                                const float* __restrict__ b2,
                                float* __restrict__ out) {
    int i = blockIdx.x * blockDim.x + threadIdx.x;
    if (i < N_NODES * C_DIM) out[i] = hroot[i] + b2[i & (C_DIM - 1)];
}

// ---------------------------------------------------------------------------
// 7) conv2 message scatter: out[dst,:] += h_all[rel, src, :] / cnt[dst,rel]
// ---------------------------------------------------------------------------
__global__ __launch_bounds__(C_DIM) void scatter2_kernel(
        const int* __restrict__ src, const int* __restrict__ dst,
        const int* __restrict__ et,  const float* __restrict__ h_all,
        const float* __restrict__ cnt, float* __restrict__ out) {
    int e = blockIdx.x;
    int f = threadIdx.x;
    int s = src[e], d = dst[e], r = et[e];
    float inv = 1.0f / cnt[d * R_REL + r];
    float v = h_all[((size_t)r * N_NODES + s) * C_DIM + f] * inv;
    atomicAdd(&out[(size_t)d * C_DIM + f], v);
}

// ---------------------------------------------------------------------------
extern "C" void kernel_launch(void* const* d_in, const int* in_sizes, int n_in,
                              void* d_out, int out_size, void* d_ws, size_t ws_size,
                              hipStream_t stream) {
    (void)in_sizes; (void)n_in; (void)out_size; (void)ws_size;

    const int*   edge_index = (const int*)d_in[0];   // [2, E]
    const int*   src   = edge_index;
    const int*   dst   = edge_index + E_EDGES;
    const int*   et    = (const int*)d_in[1];        // [E]
    const float* W1    = (const float*)d_in[2];      // [R, N, H]
    const float* root1 = (const float*)d_in[3];      // [N, H]
    const float* b1    = (const float*)d_in[4];      // [H]
    const float* W2    = (const float*)d_in[5];      // [R, H, C]
    const float* root2 = (const float*)d_in[6];      // [H, C]
    const float* b2    = (const float*)d_in[7];      // [C]
    float*       out   = (float*)d_out;              // [N, C]

    // workspace layout (floats): cnt[N*R] | h[N*H] | h_all[R*N*C] | hroot[N*C]
    float* ws    = (float*)d_ws;
    float* cnt   = ws;
    float* h     = cnt   + (size_t)N_NODES * R_REL;
    float* h_all = h     + (size_t)N_NODES * H_DIM;
    float* hroot = h_all + (size_t)R_REL * N_NODES * C_DIM;

    // 1) zero cnt + h (contiguous)
    int nz = N_NODES * R_REL + N_NODES * H_DIM;
    zero_kernel<<<(nz + 255) / 256, 256, 0, stream>>>(cnt, nz);

    // 2) edge counts per (dst, rel)
    count_kernel<<<(E_EDGES + 255) / 256, 256, 0, stream>>>(dst, et, cnt);

    // 3) conv1 scatter-mean
    scatter1_kernel<<<E_EDGES, H_DIM, 0, stream>>>(src, dst, et, W1, cnt, h);

    // 4) h = relu(h + root1 + b1)
    relu_add_kernel<<<(N_NODES * H_DIM + 255) / 256, 256, 0, stream>>>(h, root1, b1);

    // 5) WMMA GEMMs: h_all[r] = h @ W2[r] (z=0..7), hroot = h @ root2 (z=8)
    dim3 ggrid(625, C_DIM / 16, R_REL + 1);
    wmma_gemm_kernel<<<ggrid, 128, 0, stream>>>(h, W2, root2, h_all, hroot);

    // 6) out = hroot + b2
    init_out_kernel<<<(N_NODES * C_DIM + 255) / 256, 256, 0, stream>>>(hroot, b2, out);

    // 7) conv2 scatter-mean
    scatter2_kernel<<<E_EDGES, C_DIM, 0, stream>>>(src, dst, et, h_all, cnt, out);
}